// GNNSupernetwork_8942121910378
// MI455X (gfx1250) — compile-verified
//
#include <hip/hip_runtime.h>

#define NN 1000000      // nodes
#define NE 16000000     // edges
#define F  6            // feature dim (num_operations)

typedef __attribute__((ext_vector_type(2))) float v2f;  // f32 WMMA A/B: 16x4 (4x16) = 2 VGPRs
typedef __attribute__((ext_vector_type(8))) float v8f;  // f32 WMMA C/D: 16x16 = 8 VGPRs

// -------------------------------------------------------------------------
// Zero a float buffer (vectorized, NN*F divisible by 4)
// -------------------------------------------------------------------------
__global__ void k_zero(float4* __restrict__ p, int n4) {
  int i = blockIdx.x * blockDim.x + threadIdx.x;
  if (i < n4) {
    float4 z = {0.f, 0.f, 0.f, 0.f};
    p[i] = z;
  }
}

// -------------------------------------------------------------------------
// Edge scatter-add: agg[dst] += feat[src]   (segment_sum)
// One thread per edge; node arrays (24 MB) are L2-resident, atomics hit L2.
// -------------------------------------------------------------------------
__global__ void k_scatter(const float* __restrict__ feat,
                          const int*   __restrict__ src,
                          const int*   __restrict__ dst,
                          float*       __restrict__ agg) {
  int e = blockIdx.x * blockDim.x + threadIdx.x;
  if (e >= NE) return;
  int s = src[e];
  int d = dst[e];
  const float* fr = feat + (size_t)s * F;
  float*       ar = agg  + (size_t)d * F;
  float f0 = fr[0], f1 = fr[1], f2 = fr[2], f3 = fr[3], f4 = fr[4], f5 = fr[5];
  unsafeAtomicAdd(ar + 0, f0);   // global_atomic_add_f32, no return
  unsafeAtomicAdd(ar + 1, f1);
  unsafeAtomicAdd(ar + 2, f2);
  unsafeAtomicAdd(ar + 3, f3);
  unsafeAtomicAdd(ar + 4, f4);
  unsafeAtomicAdd(ar + 5, f5);
}

// -------------------------------------------------------------------------
// Dense node update via WMMA: h = agg @ Wrel^T + x @ Wroot^T + b
// One wave per 16-node tile; K=6 padded to two K=4 slices, N=6 padded to 16.
// -------------------------------------------------------------------------
__device__ __forceinline__ float ld_feat(const float* __restrict__ row, int k) {
  float v = row[(k < F) ? k : 0];          // clamped address, select below
  return (k < F) ? v : 0.f;
}
__device__ __forceinline__ float ld_w(const float* __restrict__ W, int n, int k) {
  // B[k][n] = W[n][k]  (out = in @ W^T), zero-padded outside 6x6
  float v = W[((n < F) ? n : 0) * F + ((k < F) ? k : 0)];
  return (n < F && k < F) ? v : 0.f;
}

__global__ __launch_bounds__(256)
void k_node_update(const float* __restrict__ agg,
                   const float* __restrict__ xr,
                   const float* __restrict__ Wrel,
                   const float* __restrict__ Wroot,
                   const float* __restrict__ bias,
                   float*       __restrict__ hout) {
  const int lane = threadIdx.x & 31;
  const int wv   = threadIdx.x >> 5;
  const int base = (blockIdx.x * 8 + wv) * 16;   // 16 nodes per wave
  if (base >= NN) return;                        // wave-uniform exit (EXEC stays full for WMMA)

  const int m  = lane & 15;   // A row / B,D column within tile
  const int hi = lane >> 4;   // lane half

  // ---- A tiles: rows = nodes, component v <-> K = v + 2*hi (+4 for slice 1) ----
  const float* arow = agg + (size_t)(base + m) * F;
  const float* xrow = xr  + (size_t)(base + m) * F;
  v2f aa0, aa1, ax0, ax1;
  aa0.x = ld_feat(arow, 2 * hi + 0);  aa0.y = ld_feat(arow, 2 * hi + 1);
  aa1.x = ld_feat(arow, 2 * hi + 4);  aa1.y = ld_feat(arow, 2 * hi + 5);
  ax0.x = ld_feat(xrow, 2 * hi + 0);  ax0.y = ld_feat(xrow, 2 * hi + 1);
  ax1.x = ld_feat(xrow, 2 * hi + 4);  ax1.y = ld_feat(xrow, 2 * hi + 5);

  // ---- B tiles: 4x16, component v <-> K = v + 2*hi, column n = m ----
  v2f br0, br1, bt0, bt1;
  br0.x = ld_w(Wrel,  m, 2 * hi + 0);  br0.y = ld_w(Wrel,  m, 2 * hi + 1);
  br1.x = ld_w(Wrel,  m, 2 * hi + 4);  br1.y = ld_w(Wrel,  m, 2 * hi + 5);
  bt0.x = ld_w(Wroot, m, 2 * hi + 0);  bt0.y = ld_w(Wroot, m, 2 * hi + 1);
  bt1.x = ld_w(Wroot, m, 2 * hi + 4);  bt1.y = ld_w(Wroot, m, 2 * hi + 5);

  // ---- C init: broadcast bias along rows (column n = m) ----
  float bn = (m < F) ? bias[m] : 0.f;
  v8f c;
#pragma unroll
  for (int i = 0; i < 8; ++i) c[i] = bn;

  // ---- 4 chained f32 WMMAs: D = A*B + C ----
  c = __builtin_amdgcn_wmma_f32_16x16x4_f32(false, aa0, false, br0, (short)0, c, false, false);
  c = __builtin_amdgcn_wmma_f32_16x16x4_f32(false, aa1, false, br1, (short)0, c, false, false);
  c = __builtin_amdgcn_wmma_f32_16x16x4_f32(false, ax0, false, bt0, (short)0, c, false, false);
  c = __builtin_amdgcn_wmma_f32_16x16x4_f32(false, ax1, false, bt1, (short)0, c, false, false);

  // ---- Store: D VGPR v -> row v + 8*hi, column m; keep columns 0..5 ----
  if (m < F) {
#pragma unroll
    for (int v = 0; v < 8; ++v)
      hout[(size_t)(base + v + hi * 8) * F + m] = c[v];
  }
}

// -------------------------------------------------------------------------
// softmax -> argmax -> one_hot; softmax is monotone, argmax directly
// (strict '>' preserves jnp.argmax first-max tie rule)
// -------------------------------------------------------------------------
__global__ void k_argmax_onehot(const float* __restrict__ h, float* __restrict__ out) {
  int n = blockIdx.x * blockDim.x + threadIdx.x;
  if (n >= NN) return;
  const float* r = h + (size_t)n * F;
  float bv = r[0];
  int   bi = 0;
#pragma unroll
  for (int k = 1; k < F; ++k) {
    float v = r[k];
    if (v > bv) { bv = v; bi = k; }
  }
  float* o = out + (size_t)n * F;
#pragma unroll
  for (int k = 0; k < F; ++k) o[k] = (k == bi) ? 1.f : 0.f;
}

// -------------------------------------------------------------------------
extern "C" void kernel_launch(void* const* d_in, const int* in_sizes, int n_in,
                              void* d_out, int out_size, void* d_ws, size_t ws_size,
                              hipStream_t stream) {
  const float* x      = (const float*)d_in[0];
  const int*   ei     = (const int*)  d_in[1];  // (2, NE): row 0 = src, row 1 = dst
  const float* Wrel1  = (const float*)d_in[2];
  const float* Wroot1 = (const float*)d_in[3];
  const float* b1     = (const float*)d_in[4];
  const float* Wrel2  = (const float*)d_in[5];
  const float* Wroot2 = (const float*)d_in[6];
  const float* b2     = (const float*)d_in[7];
  float* out = (float*)d_out;

  const int* src = ei;
  const int* dst = ei + NE;

  float* agg = (float*)d_ws;                 // NN*F floats (24 MB), reused per layer
  float* h1  = agg + (size_t)NN * F;         // NN*F floats
  float* h2  = h1  + (size_t)NN * F;         // NN*F floats

  const int n4 = NN * F / 4;
  dim3 blk(256);
  dim3 gZero((n4 + 255) / 256);
  dim3 gEdge((NE + 255) / 256);
  dim3 gTile(((NN / 16) + 7) / 8);           // 8 waves/block, 16 nodes/wave
  dim3 gNode((NN + 255) / 256);

  // ---- layer 1 ----
  k_zero<<<gZero, blk, 0, stream>>>((float4*)agg, n4);
  k_scatter<<<gEdge, blk, 0, stream>>>(x, src, dst, agg);
  k_node_update<<<gTile, blk, 0, stream>>>(agg, x, Wrel1, Wroot1, b1, h1);

  // ---- layer 2 ----
  k_zero<<<gZero, blk, 0, stream>>>((float4*)agg, n4);
  k_scatter<<<gEdge, blk, 0, stream>>>(h1, src, dst, agg);
  k_node_update<<<gTile, blk, 0, stream>>>(agg, h1, Wrel2, Wroot2, b2, h2);

  // ---- interpret_operations ----
  k_argmax_onehot<<<gNode, blk, 0, stream>>>(h2, out);
}